// MultiHeadAttention_27041114096054
// MI455X (gfx1250) — compile-verified
//
#include <hip/hip_runtime.h>
#include <hip/hip_bf16.h>

typedef float v2f __attribute__((ext_vector_type(2)));
typedef float v8f __attribute__((ext_vector_type(8)));
typedef int   v4i __attribute__((vector_size(16)));   // b128 payload type

#define LSTRIDE 68           // 64 floats + 4 pad (attention tiles)
#define ASTRIDE 36           // 32 floats + 4 pad (gemm A tiles)
#define S_LEN   2048
#define D_MOD   512
#define HEADS   8
#define HDIM    64
#define NEGBIG  10000.0f

#define GLB_AS __attribute__((address_space(1)))
#define LDS_AS __attribute__((address_space(3)))

#if __has_builtin(__builtin_amdgcn_global_load_async_to_lds_b128)
#define ASYNC_COPY_OK 1
#endif

#if __has_builtin(__builtin_amdgcn_s_wait_asynccnt)
#define WAIT_ASYNC(n) __builtin_amdgcn_s_wait_asynccnt(n)
#elif defined(ASYNC_COPY_OK)
#define WAIT_ASYNC(n) asm volatile("s_wait_asynccnt %0" ::"n"(n))
#else
#define WAIT_ASYNC(n)
#endif

// 16-byte global -> LDS copy: async (ASYNCcnt-tracked, no VGPR roundtrip)
// when the CDNA5 builtin exists, else plain load/store fallback.
static __device__ __forceinline__ void copy16(const float* __restrict__ g,
                                              float* l) {
#ifdef ASYNC_COPY_OK
  __builtin_amdgcn_global_load_async_to_lds_b128((GLB_AS v4i*)g,
                                                 (LDS_AS v4i*)l, 0, 0);
#else
  *(float4*)l = *(const float4*)g;
#endif
}

static __device__ __forceinline__ v8f wmma4(v2f a, v2f b, v8f c) {
  // V_WMMA_F32_16X16X4_F32 : D = A(16x4) * B(4x16) + C(16x16), all fp32
  return __builtin_amdgcn_wmma_f32_16x16x4_f32(false, a, false, b, (short)0, c,
                                               false, false);
}

// ---------------------------------------------------------------------------
// GEMM: out = relu(X[M=4096,K=512] @ W[512,512] + bias)
// Double-buffered async-to-LDS pipeline, K-tile = 32.
// HEADSPLIT=1 -> out indexed as [B,H,S,hd]; HEADSPLIT=0 -> row-major [M,N]
// grid = (N/64, M/64), block = 128 (4 waves); each wave does a 16x64 strip.
// ---------------------------------------------------------------------------
template <int HEADSPLIT>
__global__ __launch_bounds__(128)
void gemm_bias_relu_kernel(const float* __restrict__ X,
                           const float* __restrict__ W,
                           const float* __restrict__ bias,
                           float* __restrict__ out) {
  __shared__ float As[2][64 * ASTRIDE];  // [m=64][k=32] (+pad)
  __shared__ float Bs[2][32 * 64];       // [k=32][n=64]

  const int tid  = threadIdx.x;
  const int lane = tid & 31;
  const int wy   = tid >> 5;   // wave id 0..3
  const int lo   = lane & 15;  // lane within half
  const int hi   = lane >> 4;  // half-wave select
  const int mBase = blockIdx.y * 64;
  const int nBase = blockIdx.x * 64;

  // prefetch of one K-tile = 8 async ops per thread (4 for A, 4 for B)
  auto prefetch = [&](int buf, int kt) {
#pragma unroll
    for (int j = 0; j < 4; ++j) {  // A: 64 rows x 8 float4
      int idx = j * 128 + tid;
      int row = idx >> 3;
      int c4  = (idx & 7) << 2;
      copy16(&X[(size_t)(mBase + row) * D_MOD + kt + c4],
             &As[buf][row * ASTRIDE + c4]);
    }
#pragma unroll
    for (int j = 0; j < 4; ++j) {  // B: 32 rows x 16 float4
      int idx = j * 128 + tid;
      int row = idx >> 4;
      int c4  = (idx & 15) << 2;
      copy16(&W[(size_t)(kt + row) * D_MOD + nBase + c4],
             &Bs[buf][row * 64 + c4]);
    }
  };

  v8f acc[4] = {};

  prefetch(0, 0);
  for (int t = 0; t < 16; ++t) {
    if (t + 1 < 16) prefetch((t + 1) & 1, (t + 1) * 32);
    WAIT_ASYNC(8);    // tile t landed in LDS (next tile's 8 remain in flight)
    __syncthreads();  // all waves' quarters of tile t are visible

    const int buf = t & 1;
#pragma unroll
    for (int k0 = 0; k0 < 32; k0 += 4) {
      // A frag: lane (hi,lo) holds row=lo, k = k0+2*hi, k0+2*hi+1 (contiguous)
      v2f a = *(const v2f*)&As[buf][(wy * 16 + lo) * ASTRIDE + k0 + 2 * hi];
#pragma unroll
      for (int nt = 0; nt < 4; ++nt) {
        v2f b;
        b.x = Bs[buf][(k0 + 2 * hi) * 64 + nt * 16 + lo];
        b.y = Bs[buf][(k0 + 2 * hi + 1) * 64 + nt * 16 + lo];
        acc[nt] = wmma4(a, b, acc[nt]);
      }
    }
    __syncthreads();  // done reading buf before it is refilled
  }

  // epilogue: bias + relu + store (C layout: vgpr v -> row v + 8*hi, col lo)
#pragma unroll
  for (int nt = 0; nt < 4; ++nt) {
    const int n  = nBase + nt * 16 + lo;
    const float bb = bias[n];
#pragma unroll
    for (int v = 0; v < 8; ++v) {
      const int m = mBase + wy * 16 + v + 8 * hi;
      float val = acc[nt][v] + bb;
      val = val > 0.0f ? val : 0.0f;
      if (HEADSPLIT) {
        const int bidx = m >> 11;          // m / 2048
        const int s    = m & (S_LEN - 1);
        const int hh   = n >> 6;           // n / 64
        const int dd   = n & (HDIM - 1);
        out[(((size_t)(bidx * HEADS + hh) * S_LEN) + s) * HDIM + dd] = val;
      } else {
        out[(size_t)m * D_MOD + n] = val;
      }
    }
  }
}

// ---------------------------------------------------------------------------
// Flash-style causal attention per (b,h): softmax(QK^T*scale + masks) @ V
// Q,K,V in [B*H, S, 64]; grid = (S/64 q-tiles, B*H); block = 128 (4 waves).
// Each wave owns 16 q rows; K loop streams 64-row K/V tiles through LDS
// via async-to-LDS copies.
// ---------------------------------------------------------------------------
__global__ __launch_bounds__(128)
void attn_kernel(const float* __restrict__ Q, const float* __restrict__ K,
                 const float* __restrict__ V, const int* __restrict__ mask,
                 float* __restrict__ O) {
  __shared__ float Ks[64 * LSTRIDE];
  __shared__ float Vs[64 * LSTRIDE];
  __shared__ float Ps[64 * LSTRIDE];
  __shared__ float kadd[64];

  const int tid  = threadIdx.x;
  const int lane = tid & 31;
  const int wy   = tid >> 5;
  const int lo   = lane & 15;
  const int hi   = lane >> 4;
  const int bh   = blockIdx.y;
  const int b    = bh >> 3;        // / HEADS
  const int h    = bh & (HEADS - 1);
  const int qBase = blockIdx.x * 64;

  const float* Qp = Q + (size_t)bh * S_LEN * HDIM;
  const float* Kp = K + (size_t)bh * S_LEN * HDIM;
  const float* Vp = V + (size_t)bh * S_LEN * HDIM;

  // Q fragments held in registers, pre-scaled by 1/sqrt(hd)=0.125
  v2f qf[16];
  {
    const int qrow = qBase + wy * 16 + lo;
#pragma unroll
    for (int ks = 0; ks < 16; ++ks) {
      v2f t = *(const v2f*)&Qp[(size_t)qrow * HDIM + ks * 4 + 2 * hi];
      t.x *= 0.125f;
      t.y *= 0.125f;
      qf[ks] = t;
    }
  }

  float mrow[8], lrow[8];
#pragma unroll
  for (int v = 0; v < 8; ++v) { mrow[v] = -3.0e38f; lrow[v] = 0.0f; }
  v8f o[4] = {};

  for (int kt = 0; kt <= (int)blockIdx.x; ++kt) {
    const int kBase = kt * 64;
#pragma unroll
    for (int j = 0; j < 8; ++j) {
      int idx = j * 128 + tid;
      int row = idx >> 4;
      int c4  = (idx & 15) << 2;
      copy16(&Kp[(size_t)(kBase + row) * HDIM + c4], &Ks[row * LSTRIDE + c4]);
      copy16(&Vp[(size_t)(kBase + row) * HDIM + c4], &Vs[row * LSTRIDE + c4]);
    }
    if (tid < 64)
      kadd[tid] = -NEGBIG * (1.0f - (float)mask[b * S_LEN + kBase + tid]);
    WAIT_ASYNC(0);
    __syncthreads();

    // scores S = (Q*scale) @ K^T ; B-frag = contiguous float2 from K row n
    v8f s[4] = {};
#pragma unroll
    for (int k0 = 0; k0 < 16; ++k0) {
#pragma unroll
      for (int nt = 0; nt < 4; ++nt) {
        v2f bfr = *(const v2f*)&Ks[(nt * 16 + lo) * LSTRIDE + k0 * 4 + 2 * hi];
        s[nt] = wmma4(qf[k0], bfr, s[nt]);
      }
    }

    const bool diag = (kt == (int)blockIdx.x);

    // streaming softmax; each lane owns rows v + 8*hi of its wave's 16 rows
#pragma unroll
    for (int v = 0; v < 8; ++v) {
      const int qi = qBase + wy * 16 + v + 8 * hi;
      float xv[4];
      float xm = -3.0e38f;
#pragma unroll
      for (int nt = 0; nt < 4; ++nt) {
        const int ki = kBase + nt * 16 + lo;
        float x = s[nt][v] + kadd[nt * 16 + lo];
        if (diag && ki > qi) x -= NEGBIG;   // additive causal mask (as ref)
        xv[nt] = x;
        xm = fmaxf(xm, x);
      }
#pragma unroll
      for (int mk = 1; mk < 16; mk <<= 1)
        xm = fmaxf(xm, __shfl_xor(xm, mk, 32));
      const float mn = fmaxf(mrow[v], xm);
      const float f  = __expf(mrow[v] - mn);
      mrow[v] = mn;
      float psum = 0.0f;
#pragma unroll
      for (int nt = 0; nt < 4; ++nt) {
        const float p = __expf(xv[nt] - mn);
        psum += p;
        Ps[(wy * 16 + v + 8 * hi) * LSTRIDE + nt * 16 + lo] = p;  // wave-local
      }
#pragma unroll
      for (int mk = 1; mk < 16; mk <<= 1)
        psum += __shfl_xor(psum, mk, 32);
      lrow[v] = lrow[v] * f + psum;
#pragma unroll
      for (int nt = 0; nt < 4; ++nt) o[nt][v] *= f;
    }

    // O += P @ V  (A-frag re-read from LDS converts C-layout -> A-layout)
#pragma unroll
    for (int k0 = 0; k0 < 16; ++k0) {
      v2f a = *(const v2f*)&Ps[(wy * 16 + lo) * LSTRIDE + k0 * 4 + 2 * hi];
#pragma unroll
      for (int nt = 0; nt < 4; ++nt) {
        v2f bfr;
        bfr.x = Vs[(k0 * 4 + 2 * hi) * LSTRIDE + nt * 16 + lo];
        bfr.y = Vs[(k0 * 4 + 2 * hi + 1) * LSTRIDE + nt * 16 + lo];
        o[nt] = wmma4(a, bfr, o[nt]);
      }
    }
    __syncthreads();
  }

  // finalize: divide by row sums, write merged-head layout [B,S, h*64+d]
#pragma unroll
  for (int v = 0; v < 8; ++v) {
    const float inv = 1.0f / lrow[v];
    const int srow = qBase + wy * 16 + v + 8 * hi;
#pragma unroll
    for (int nt = 0; nt < 4; ++nt) {
      O[((size_t)(b * S_LEN + srow)) * D_MOD + h * HDIM + nt * 16 + lo] =
          o[nt][v] * inv;
    }
  }
}

// ---------------------------------------------------------------------------
extern "C" void kernel_launch(void* const* d_in, const int* in_sizes, int n_in,
                              void* d_out, int out_size, void* d_ws,
                              size_t ws_size, hipStream_t stream) {
  const float* x    = (const float*)d_in[0];
  const int*   mask = (const int*)d_in[1];
  const float* Wq   = (const float*)d_in[2];
  const float* bq   = (const float*)d_in[3];
  const float* Wk   = (const float*)d_in[4];
  const float* bk   = (const float*)d_in[5];
  const float* Wv   = (const float*)d_in[6];
  const float* bv   = (const float*)d_in[7];
  const float* Wo   = (const float*)d_in[8];
  const float* bo   = (const float*)d_in[9];

  float* ws  = (float*)d_ws;
  const size_t perT = (size_t)2 * HEADS * S_LEN * HDIM;  // 2,097,152 floats
  float* Qw = ws;
  float* Kw = ws + perT;
  float* Vw = ws + 2 * perT;
  float* Ow = ws + 3 * perT;

  dim3 gGemm(D_MOD / 64, (2 * S_LEN) / 64);  // (8, 64)
  dim3 blk(128);

  gemm_bias_relu_kernel<1><<<gGemm, blk, 0, stream>>>(x, Wq, bq, Qw);
  gemm_bias_relu_kernel<1><<<gGemm, blk, 0, stream>>>(x, Wk, bk, Kw);
  gemm_bias_relu_kernel<1><<<gGemm, blk, 0, stream>>>(x, Wv, bv, Vw);

  dim3 gAttn(S_LEN / 64, 2 * HEADS);  // (32, 16)
  attn_kernel<<<gAttn, blk, 0, stream>>>(Qw, Kw, Vw, mask, Ow);

  gemm_bias_relu_kernel<0><<<gGemm, blk, 0, stream>>>(Ow, Wo, bo,
                                                      (float*)d_out);
}